// User_Graph_sample_8297876816694
// MI455X (gfx1250) — compile-verified
//
#include <hip/hip_runtime.h>

typedef __attribute__((ext_vector_type(2))) float v2f;
typedef __attribute__((ext_vector_type(8))) float v8f;

#define NUM_USER 1000000
#define KNN 10
#define DIM 32

// One wave32 handles 16 users. out[u,d] = sum_k w[u,k] * F[idx[u,k], d]
// realized as 80x V_WMMA_F32_16X16X4_F32 per tile via diag(w) x gathered-rows.
__global__ __launch_bounds__(128) void user_graph_wmma_kernel(
    const float* __restrict__ features,              // [NUM_USER, DIM] f32
    const unsigned long long* __restrict__ graph,    // [NUM_USER, KNN] i64
    const float* __restrict__ user_matrix,           // [NUM_USER, KNN] f32
    float* __restrict__ out)                         // [NUM_USER, DIM] f32
{
    const int wave = blockIdx.x * (blockDim.x >> 5) + (threadIdx.x >> 5);
    const int lane = threadIdx.x & 31;
    const int m = lane & 15;   // N index (dim within half) / A-row index
    const int h = lane >> 4;   // half-wave: selects K-slot pair {0,1} vs {2,3}

    // u0 is wave-uniform; make it provably uniform.
    const int u0 = __builtin_amdgcn_readfirstlane(wave * 16);

    // Each lane only ever contributes its OWN user's weight to the diag-A:
    // preload w[u0+m, 0..9] into registers (branch-free selects later).
    float wk[KNN];
    {
        const unsigned wbase = (unsigned)(u0 + m) * KNN;
        #pragma unroll
        for (int k = 0; k < KNN; ++k) wk[k] = user_matrix[wbase + k];
    }

    v8f acc0 = {}; // dims 0..15 for users u0..u0+15
    v8f acc1 = {}; // dims 16..31

    #pragma unroll
    for (int c = 0; c < 4; ++c) {            // sub-group of 4 users
        const bool selx = (m == 4 * c + 2 * h);     // lane owns A[m, 2h]
        const bool sely = (m == 4 * c + 2 * h + 1); // lane owns A[m, 2h+1]
        // index rows for the two users this half-wave feeds into B
        const unsigned gx = (unsigned)(u0 + 4 * c + 2 * h) * KNN; // user 4c+2h
        const unsigned gy = gx + KNN;                             // user 4c+2h+1

        #pragma unroll
        for (int k = 0; k < KNN; ++k) {      // neighbor slot
            // A = 16x4 slice of diag(w[:,k]); pure v_cndmask on registers.
            v2f a;
            a.x = selx ? wk[k] : 0.0f;
            a.y = sely ? wk[k] : 0.0f;

            // Gather row ids: 32-bit unsigned index into u64 array ->
            // global_load_b64 vdst, v_idx, s[base] scale_offset (GVS mode).
            const unsigned rx = (unsigned)graph[gx + k];
            const unsigned ry = (unsigned)graph[gy + k];
            const unsigned ox = rx * DIM + m;   // 32-bit offsets -> GVS mode
            const unsigned oy = ry * DIM + m;

            v2f b0, b1;
            b0.x = features[ox];      b0.y = features[oy];      // dims 0..15
            b1.x = features[ox + 16]; b1.y = features[oy + 16]; // dims 16..31

            acc0 = __builtin_amdgcn_wmma_f32_16x16x4_f32(
                false, a, false, b0, (short)0, acc0, false, false);
            acc1 = __builtin_amdgcn_wmma_f32_16x16x4_f32(
                false, a, false, b1, (short)0, acc1, false, false);
        }
    }

    // C/D layout: VGPR j, lanes<16 -> user u0+j, lanes>=16 -> user u0+j+8; N = m.
    const unsigned obase = (unsigned)(u0 + 8 * h) * DIM + m;
    #pragma unroll
    for (int j = 0; j < 8; ++j) {
        out[obase + j * DIM]      = acc0[j];
        out[obase + j * DIM + 16] = acc1[j];
    }
}

extern "C" void kernel_launch(void* const* d_in, const int* in_sizes, int n_in,
                              void* d_out, int out_size, void* d_ws, size_t ws_size,
                              hipStream_t stream) {
    const float*              features    = (const float*)d_in[0];              // [U, 32] f32
    const unsigned long long* graph       = (const unsigned long long*)d_in[1]; // [U, 10] i64
    const float*              user_matrix = (const float*)d_in[2];              // [U, 10] f32
    float*                    out         = (float*)d_out;                      // [U, 32] f32

    // 1M users / 16 per wave = 62500 waves; 4 waves (128 thr) per block.
    const int tiles  = NUM_USER / 16;     // 62500
    const int blocks = tiles / 4;         // 15625 (exact)
    user_graph_wmma_kernel<<<blocks, 128, 0, stream>>>(
        features, graph, user_matrix, out);
}